// DropBlock_31791347925029
// MI455X (gfx1250) — compile-verified
//
#include <hip/hip_runtime.h>
#include <stdint.h>

// DropBlock, BLOCK_SIZE=5, x:[128,64,56,56] f32, u:[128,64,52,52] f32, gamma:[1] f32
#define BSZ 5
#define HH 56
#define WW 56
#define HM 52          // H - BSZ + 1
#define WM 52
#define PLANE (HH * WW)      // 3136
#define UPLANE (HM * WM)     // 2704
#define UVEC4 (UPLANE / 4)   // 676

// Exact pointee type expected by the async-LDS builtin (from hipcc diagnostic):
// 'int __attribute__((vector_size(16)))' in global (AS1) / LDS (AS3) space.
typedef int v4i __attribute__((vector_size(4 * sizeof(int))));
typedef __attribute__((address_space(1))) v4i g_v4i;
typedef __attribute__((address_space(3))) v4i l_v4i;

// ---- CDNA5 async global->LDS DMA (ASYNCcnt path) ----------------------------
__device__ __forceinline__ void async_copy_b128(const void* gsrc, void* ldst) {
#if __has_builtin(__builtin_amdgcn_global_load_async_to_lds_b128)
  __builtin_amdgcn_global_load_async_to_lds_b128(
      (g_v4i*)gsrc, (l_v4i*)ldst, /*offset=*/0, /*cpol=*/0);
#else
  unsigned loff = (unsigned)(unsigned long long)(l_v4i*)ldst;
  unsigned long long ga = (unsigned long long)gsrc;
  asm volatile("global_load_async_to_lds_b128 %0, %1, off"
               :: "v"(loff), "v"(ga) : "memory");
#endif
}

__device__ __forceinline__ void wait_async_zero() {
#if __has_builtin(__builtin_amdgcn_s_wait_asynccnt)
  __builtin_amdgcn_s_wait_asynccnt(0);
#else
  asm volatile("s_wait_asynccnt 0" ::: "memory");
#endif
}

// ---- Pass 0: zero the global keep-counter ----------------------------------
__global__ void zero_counter_kernel(unsigned* __restrict__ c) {
  if (threadIdx.x == 0 && blockIdx.x == 0) *c = 0u;
}

// ---- Pass 1: per-plane seed mask -> 5x5 dilation -> per-row keep bitmasks ---
// One 256-thread workgroup per (b,c) plane. u-plane is DMA'd into LDS.
__global__ __launch_bounds__(256) void dropblock_mask_kernel(
    const float* __restrict__ u, const float* __restrict__ gamma,
    unsigned long long* __restrict__ row_masks,   // [BC * HH]
    unsigned* __restrict__ count) {
  __shared__ float su[UPLANE];                 // 10816 B
  __shared__ unsigned long long hdil[HM];      // horizontally dilated rows
  __shared__ unsigned scount;

  const int plane = blockIdx.x;
  const int t = threadIdx.x;
  const float* up = u + (size_t)plane * UPLANE;

  // Async DMA the 52x52 tile into LDS: 676 x 16B transfers, 256 lanes/iter.
  for (int i = t; i < UVEC4; i += 256) {
    async_copy_b128(up + i * 4, &su[i * 4]);
  }
  if (t == 0) scount = 0u;
  wait_async_zero();       // this wave's ASYNCcnt == 0 (LDS writes landed)
  __syncthreads();         // all waves' DMA visible workgroup-wide

  const float g = gamma[0];

  // Row seed bits + horizontal dilation: seed at col w' covers cols w'..w'+4.
  if (t < HM) {
    const float* row = &su[t * WM];
    unsigned long long m = 0ull;
#pragma unroll
    for (int j = 0; j < WM; ++j)
      m |= (unsigned long long)(row[j] < g ? 1u : 0u) << j;
    hdil[t] = m | (m << 1) | (m << 2) | (m << 3) | (m << 4);
  }
  __syncthreads();

  // Vertical dilation: output row h sees seed rows [h-4, h] clipped to [0,51].
  if (t < HH) {
    int lo = t - (BSZ - 1); if (lo < 0) lo = 0;
    int hi = t;             if (hi > HM - 1) hi = HM - 1;
    unsigned long long v = 0ull;
    for (int h = lo; h <= hi; ++h) v |= hdil[h];
    unsigned long long keep = (~v) & ((1ull << WW) - 1ull);  // 56 valid bits
    row_masks[(size_t)plane * HH + t] = keep;
    atomicAdd(&scount, (unsigned)__popcll(keep));
  }
  __syncthreads();
  if (t == 0) atomicAdd(count, scount);   // integer atomics: deterministic
}

// ---- Pass 2: out = keep ? x * (N / count) : 0, float4 streaming ------------
__global__ __launch_bounds__(256) void dropblock_apply_kernel(
    const float* __restrict__ x,
    const unsigned long long* __restrict__ row_masks,
    const unsigned* __restrict__ count,
    float* __restrict__ out, int nvec, float countM) {
  int i = blockIdx.x * blockDim.x + threadIdx.x;
  if (i >= nvec) return;

  // Prefetch the x stream ahead (global_prefetch_b8; speculative OOB is ok).
  __builtin_prefetch((const char*)x + (size_t)i * 16 + 32768, 0, 0);

  const float scale = countM / (float)(*count);

  int flat = i * 4;                 // 56 % 4 == 0 -> vec4 never crosses a row
  int plane = flat / PLANE;
  int rem = flat - plane * PLANE;
  int h = rem / WW;
  int w = rem - h * WW;

  float4 xv = ((const float4*)x)[i];
  unsigned long long m = row_masks[(size_t)plane * HH + h];
  unsigned b = (unsigned)(m >> w);

  float4 o;
  o.x = (b & 1u) ? xv.x * scale : 0.0f;
  o.y = (b & 2u) ? xv.y * scale : 0.0f;
  o.z = (b & 4u) ? xv.z * scale : 0.0f;
  o.w = (b & 8u) ? xv.w * scale : 0.0f;
  ((float4*)out)[i] = o;
}

extern "C" void kernel_launch(void* const* d_in, const int* in_sizes, int n_in,
                              void* d_out, int out_size, void* d_ws, size_t ws_size,
                              hipStream_t stream) {
  const float* x     = (const float*)d_in[0];
  const float* u     = (const float*)d_in[1];
  const float* gamma = (const float*)d_in[2];
  float* out = (float*)d_out;

  // Workspace layout: [0,4) counter, [256, 256 + BC*56*8) row masks.
  unsigned* counter = (unsigned*)d_ws;
  unsigned long long* row_masks = (unsigned long long*)((char*)d_ws + 256);

  const int BC = in_sizes[1] / UPLANE;   // 128*64 = 8192 planes
  const int nvec = out_size / 4;         // float4 count

  zero_counter_kernel<<<1, 32, 0, stream>>>(counter);
  dropblock_mask_kernel<<<BC, 256, 0, stream>>>(u, gamma, row_masks, counter);
  dropblock_apply_kernel<<<(nvec + 255) / 256, 256, 0, stream>>>(
      x, row_masks, counter, out, nvec, (float)out_size);
}